// NestedDropoutUniformLayer_88476326298285
// MI455X (gfx1250) — compile-verified
//
#include <hip/hip_runtime.h>
#include <hip/hip_bf16.h>

// Nested dropout mask: out[n, j] = in[n, j] * (j < b[n])
// Shapes: in [16384, 4096] fp32, b [16384] int32, out [16384, 4096] fp32.
//
// Pure streaming op: ~512 MiB total traffic, ~22 us floor at 23.3 TB/s HBM.
// Strategy: 1 row per 256-thread block (8 wave32s), 4 fully-coalesced
// float4 (B128) chunks per thread, non-temporal load+store (single-touch
// working set >> 192 MB L2), scalar-broadcast cutoff per row.

typedef __attribute__((ext_vector_type(4))) float v4f;

#define K_DIM 4096
#define BATCH 16384
#define THREADS 256
// float4 elements per row
#define V_PER_ROW (K_DIM / 4)          // 1024
#define CHUNKS (V_PER_ROW / THREADS)   // 4

__global__ __launch_bounds__(THREADS)
void nested_dropout_kernel(const float* __restrict__ in,
                           const int* __restrict__ b,
                           float* __restrict__ out) {
    const int row = blockIdx.x;
    // Block-uniform -> s_load_b32 + SALU broadcast, no per-lane VMEM.
    const int cutoff = b[row];

    const v4f* __restrict__ src = (const v4f*)(in  + (size_t)row * K_DIM);
    v4f* __restrict__       dst = (v4f*)      (out + (size_t)row * K_DIM);

#pragma unroll
    for (int c = 0; c < CHUNKS; ++c) {
        const int v = c * THREADS + threadIdx.x;  // float4 index in row
        const int j = v << 2;                     // column of first element

        // Single-touch stream: non-temporal load (TH=NT), don't pollute L2.
        v4f x = __builtin_nontemporal_load(src + v);

        x.x = (j + 0 < cutoff) ? x.x : 0.0f;
        x.y = (j + 1 < cutoff) ? x.y : 0.0f;
        x.z = (j + 2 < cutoff) ? x.z : 0.0f;
        x.w = (j + 3 < cutoff) ? x.w : 0.0f;

        // Non-temporal store: write-stream the 256 MiB output.
        __builtin_nontemporal_store(x, dst + v);
    }
}

extern "C" void kernel_launch(void* const* d_in, const int* in_sizes, int n_in,
                              void* d_out, int out_size, void* d_ws, size_t ws_size,
                              hipStream_t stream) {
    (void)in_sizes; (void)n_in; (void)out_size; (void)d_ws; (void)ws_size;

    const float* in = (const float*)d_in[0];
    const int*   b  = (const int*)d_in[1];
    float*       out = (float*)d_out;

    dim3 grid(BATCH);   // one row per workgroup
    dim3 block(THREADS);
    nested_dropout_kernel<<<grid, block, 0, stream>>>(in, b, out);
}